// SFM_61641370632446
// MI455X (gfx1250) — compile-verified
//
#include <hip/hip_runtime.h>
#include <hip/hip_bf16.h>
#include <math.h>

typedef __attribute__((ext_vector_type(2))) float v2f;
typedef __attribute__((ext_vector_type(8))) float v8f;
typedef int vsi4 __attribute__((__vector_size__(16)));

#define DIMD    1024
#define HID     4096
#define NTOK    6
#define NSEQ    4096
#define NB      64
#define NSEG    8
#define SEGROWS 512
#define CHUNK   128
#define CHUNKP  132      /* padded row stride for p tile: 132 % 64 == 4 -> conflict-free */
#define QSTRIDE 1032     /* padded row stride for staged queries */
#define ASTRIDE 516      /* padded row stride for staged MLP A chunks (K=512) */
#define LN_EPSF 1e-5f

// ---------------------------------------------------------------------------
// gfx1250 async global->LDS staging (guarded so compilation can never break)
// Signature (from hipcc diagnostic): (v4i __device__* src, v4i __shared__* dst,
//                                     imm int offset, imm int cpol)
// ---------------------------------------------------------------------------
#if defined(__has_builtin)
#if __has_builtin(__builtin_amdgcn_global_load_async_to_lds_b128) && \
    __has_builtin(__builtin_amdgcn_s_wait_asynccnt)
#define USE_ASYNC_LDS 1
#endif
#endif
#ifndef USE_ASYNC_LDS
#define USE_ASYNC_LDS 0
#endif

#if USE_ASYNC_LDS
#define ASYNC_CP16(dstLds, srcGlb)                                   \
  __builtin_amdgcn_global_load_async_to_lds_b128(                    \
      (__attribute__((address_space(1))) vsi4*)(srcGlb),             \
      (__attribute__((address_space(3))) vsi4*)(dstLds), 0, 0)
#define ASYNC_WAIT() __builtin_amdgcn_s_wait_asynccnt(0)
#else
#define ASYNC_CP16(dstLds, srcGlb) (*(float4*)(dstLds) = *(const float4*)(srcGlb))
#define ASYNC_WAIT() ((void)0)
#endif

static __device__ __forceinline__ v8f wmma4(v2f a, v2f b, v8f c) {
  return __builtin_amdgcn_wmma_f32_16x16x4_f32(false, a, false, b, (short)0, c,
                                               false, false);
}

// ---------------------------------------------------------------------------
// Fused attention pooling with online softmax.
// Block = 8 waves, handles rows [seg*512, seg*512+512) of batch b.
// Per 128-row chunk:
//   phase A : wave w computes 16(t)x16(n) score tile (full K=1024) via WMMA,
//             x streamed from HBM once.
//   bookkeeping: running row-max m, rescale c=exp(m_old-m_new), p tile -> LDS.
//   phase B : wave w owns a 128-wide d-slice; acc = acc*c + P(16x4) * X(4x16),
//             x chunk re-read from L2.
// Emits per-(b,seg) partials: pacc[6][1024], pm[16], pl[16].
// ---------------------------------------------------------------------------
__global__ __launch_bounds__(256) void k_attnpool(const float* __restrict__ x,
                                                  const float* __restrict__ q,
                                                  float* __restrict__ pacc,
                                                  float* __restrict__ pm,
                                                  float* __restrict__ pl) {
  const int b    = blockIdx.x;
  const int seg  = blockIdx.y;
  const int wave = threadIdx.x >> 5;
  const int lane = threadIdx.x & 31;
  const int half = lane & 15;
  const int hi   = lane >> 4;
  const int khalf = hi * 2;

  __shared__ __align__(16) float qs[NTOK * QSTRIDE];   // staged queries (padded)
  __shared__ __align__(16) float sp[16 * CHUNKP];      // p tile (padded)
  __shared__ float wmax[8][16];
  __shared__ float wsum[8][16];
  __shared__ float smnew[16];
  __shared__ float sc[16];
  __shared__ float sm[16];
  __shared__ float sl[16];

  // stage queries (6 x 1024) into padded LDS rows via async-to-LDS
  for (int j = threadIdx.x; j < NTOK * 256; j += 256) {
    const int row = j >> 8;
    const int c4  = (j & 255) << 2;
    ASYNC_CP16(qs + row * QSTRIDE + c4, q + (size_t)row * DIMD + c4);
  }
  ASYNC_WAIT();
  if (threadIdx.x < 16) {
    sm[threadIdx.x] = -1e30f;
    sl[threadIdx.x] = 0.f;
  }
  __syncthreads();

  const v8f vzero = {};
  v8f acc[8];
#pragma unroll
  for (int dt = 0; dt < 8; ++dt) acc[dt] = vzero;

  const int qrow = (half < NTOK) ? half : 0;  // pad rows duplicate token 0

  for (int c = 0; c < SEGROWS / CHUNK; ++c) {
    const int nbase = seg * SEGROWS + c * CHUNK;

    // ---------------- phase A: score tile for n-slice of this wave ----------
    const int ncol = nbase + wave * 16 + half;
    const float* xrp = x + ((size_t)b * NSEQ + ncol) * DIMD + khalf;
    const float* qrp = qs + (size_t)qrow * QSTRIDE + khalf;

    v8f sacc = vzero;
    for (int k = 0; k < DIMD; k += 4) {
      if ((k & 255) == 0) __builtin_prefetch(xrp + k + 1024, 0, 0);
      v2f a  = *(const v2f*)(qrp + k);
      v2f bb = *(const v2f*)(xrp + k);
      sacc = wmma4(a, bb, sacc);
    }
#pragma unroll
    for (int i = 0; i < 8; ++i) sacc[i] *= 0.03125f;  // 1/sqrt(1024)

    // per-wave row max over the 16 columns (reduce within each 16-lane group)
#pragma unroll
    for (int i = 0; i < 8; ++i) {
      float v = sacc[i];
      v = fmaxf(v, __shfl_xor(v, 1, 32));
      v = fmaxf(v, __shfl_xor(v, 2, 32));
      v = fmaxf(v, __shfl_xor(v, 4, 32));
      v = fmaxf(v, __shfl_xor(v, 8, 32));
      if (half == 0) wmax[wave][i + 8 * hi] = v;
    }
    __syncthreads();

    // new running max + rescale factor (one thread per row)
    if (threadIdx.x < 16) {
      const int t = threadIdx.x;
      float mo = sm[t];
      float mn = mo;
#pragma unroll
      for (int w = 0; w < 8; ++w) mn = fmaxf(mn, wmax[w][t]);
      smnew[t] = mn;
      sc[t] = __expf(mo - mn);
      sm[t] = mn;
    }
    __syncthreads();

    // p = exp(s - m_new): store padded tile + per-wave row sums
#pragma unroll
    for (int i = 0; i < 8; ++i) {
      const int row = i + 8 * hi;
      float pv = __expf(sacc[i] - smnew[row]);
      sp[row * CHUNKP + wave * 16 + half] = pv;
      float sv = pv;
      sv += __shfl_xor(sv, 1, 32);
      sv += __shfl_xor(sv, 2, 32);
      sv += __shfl_xor(sv, 4, 32);
      sv += __shfl_xor(sv, 8, 32);
      if (half == 0) wsum[wave][row] = sv;
    }
    __syncthreads();

    if (threadIdx.x < 16) {
      const int t = threadIdx.x;
      float s = 0.f;
#pragma unroll
      for (int w = 0; w < 8; ++w) s += wsum[w][t];
      sl[t] = sl[t] * sc[t] + s;
    }

    // ---------------- phase B: acc = acc*c + P * X (chunk re-read from L2) --
    float cv[8];
#pragma unroll
    for (int i = 0; i < 8; ++i) cv[i] = sc[i + 8 * hi];
#pragma unroll
    for (int dt = 0; dt < 8; ++dt)
#pragma unroll
      for (int i = 0; i < 8; ++i) acc[dt][i] *= cv[i];

    const int d0 = wave * 128 + half;
    for (int kk = 0; kk < CHUNK; kk += 4) {
      v2f a = *(const v2f*)(&sp[half * CHUNKP + kk + khalf]);
      const float* xb = x + ((size_t)b * NSEQ + nbase + kk + khalf) * DIMD + d0;
#pragma unroll
      for (int dt = 0; dt < 8; ++dt) {
        v2f bb;
        bb.x = xb[dt * 16];
        bb.y = xb[DIMD + dt * 16];
        acc[dt] = wmma4(a, bb, acc[dt]);
      }
    }
    __syncthreads();  // protect sp/sc before next chunk rewrites them
  }

  // write partial results (valid token rows live in VGPRs 0..5 of lanes 0..15)
  if (lane < 16) {
    float* pp = pacc + ((size_t)(b * NSEG + seg) * NTOK) * DIMD + wave * 128 + lane;
#pragma unroll
    for (int dt = 0; dt < 8; ++dt)
#pragma unroll
      for (int t = 0; t < NTOK; ++t)
        pp[(size_t)t * DIMD + dt * 16] = acc[dt][t];
  }
  if (threadIdx.x < 16) {
    pm[(size_t)(b * NSEG + seg) * 16 + threadIdx.x] = sm[threadIdx.x];
    pl[(size_t)(b * NSEG + seg) * 16 + threadIdx.x] = sl[threadIdx.x];
  }
}

// ---------------------------------------------------------------------------
// Combine per-segment partials: pooled = (sum_s e^{m_s-M} acc_s) / (sum_s e^{m_s-M} l_s)
// ---------------------------------------------------------------------------
__global__ __launch_bounds__(256) void k_combine(const float* __restrict__ pacc,
                                                 const float* __restrict__ pm,
                                                 const float* __restrict__ pl,
                                                 float* __restrict__ pooled) {
  const int bt = blockIdx.x;           // 0..383
  const int b = bt / NTOK, t = bt % NTOK;

  float M = -1e30f;
#pragma unroll
  for (int s = 0; s < NSEG; ++s)
    M = fmaxf(M, pm[(size_t)(b * NSEG + s) * 16 + t]);
  float w[NSEG];
  float L = 0.f;
#pragma unroll
  for (int s = 0; s < NSEG; ++s) {
    float e = __expf(pm[(size_t)(b * NSEG + s) * 16 + t] - M);
    w[s] = e;
    L += e * pl[(size_t)(b * NSEG + s) * 16 + t];
  }
  const float invL = 1.0f / L;

  for (int d = threadIdx.x; d < DIMD; d += 256) {
    float v = 0.f;
#pragma unroll
    for (int s = 0; s < NSEG; ++s)
      v += w[s] * pacc[((size_t)(b * NSEG + s) * NTOK + t) * DIMD + d];
    pooled[((size_t)b * NTOK + t) * DIMD + d] = v * invL;
  }
}

// ---------------------------------------------------------------------------
// LayerNorm per row (384 rows of 1024).
// ---------------------------------------------------------------------------
__global__ __launch_bounds__(256) void k_ln(const float* __restrict__ pooled,
                                            const float* __restrict__ gamma,
                                            const float* __restrict__ beta,
                                            float* __restrict__ hn) {
  const int r = blockIdx.x;
  const float* in = pooled + (size_t)r * DIMD;
  float* outp = hn + (size_t)r * DIMD;
  const int tid = threadIdx.x;

  __shared__ float rs[8], rs2[8];
  float vals[4];
  float sum = 0.f, sum2 = 0.f;
#pragma unroll
  for (int i = 0; i < 4; ++i) {
    float v = in[tid + i * 256];
    vals[i] = v;
    sum += v;
    sum2 += v * v;
  }
  for (int o = 16; o; o >>= 1) {
    sum  += __shfl_xor(sum, o, 32);
    sum2 += __shfl_xor(sum2, o, 32);
  }
  if ((tid & 31) == 0) { rs[tid >> 5] = sum; rs2[tid >> 5] = sum2; }
  __syncthreads();
  float ts  = (rs[0] + rs[1]) + (rs[2] + rs[3]) + (rs[4] + rs[5]) + (rs[6] + rs[7]);
  float ts2 = (rs2[0] + rs2[1]) + (rs2[2] + rs2[3]) + (rs2[4] + rs2[5]) + (rs2[6] + rs2[7]);
  float mu  = ts * (1.0f / DIMD);
  float var = ts2 * (1.0f / DIMD) - mu * mu;
  float inv = rsqrtf(var + LN_EPSF);
#pragma unroll
  for (int i = 0; i < 4; ++i) {
    int c = tid + i * 256;
    outp[c] = (vals[i] - mu) * inv * gamma[c] + beta[c];
  }
}

// ---------------------------------------------------------------------------
// Kernel: h1 = GELU(hn @ w1 + b1)   M=384 N=4096 K=1024  (WMMA f32)
// A-tile (16 x 512 K-chunk) staged in LDS via async-to-LDS, shared by 8 waves.
// ---------------------------------------------------------------------------
__global__ __launch_bounds__(256) void k_mlp1(const float* __restrict__ hn,
                                              const float* __restrict__ w1,
                                              const float* __restrict__ b1,
                                              float* __restrict__ h1) {
  const int mt   = blockIdx.x;             // 0..23
  const int wave = threadIdx.x >> 5;
  const int lane = threadIdx.x & 31;
  const int half = lane & 15;
  const int m0   = mt * 16;
  const int n0   = (blockIdx.y * 8 + wave) * 16 + half;
  const int khalf = (lane >> 4) * 2;

  __shared__ __align__(16) float As[16 * ASTRIDE];  // 16 rows x 512 (padded)

  v8f acc = {};
  for (int kc = 0; kc < DIMD; kc += 512) {
    __syncthreads();  // previous chunk fully consumed
    for (int j = threadIdx.x; j < 2048; j += 256) {
      const int row = j >> 7;
      const int c4  = (j & 127) << 2;
      ASYNC_CP16(As + row * ASTRIDE + c4,
                 hn + (size_t)(m0 + row) * DIMD + kc + c4);
    }
    ASYNC_WAIT();
    __syncthreads();

    const float* arp = As + (size_t)half * ASTRIDE + khalf;
    const float* brp = w1 + (size_t)(kc + khalf) * HID + n0;
    for (int k = 0; k < 512; k += 4) {
      v2f a = *(const v2f*)(arp + k);
      v2f bb;
      bb.x = brp[(size_t)k * HID];
      bb.y = brp[(size_t)(k + 1) * HID];
      acc = wmma4(a, bb, acc);
    }
  }

  const float bias = b1[n0];
  float* op = h1 + (size_t)(m0 + 8 * (lane >> 4)) * HID + n0;
#pragma unroll
  for (int i = 0; i < 8; ++i) {
    float v = acc[i] + bias;
    v = 0.5f * v * (1.0f + erff(v * 0.70710678118654752f));  // exact GELU
    op[(size_t)i * HID] = v;
  }
}

// ---------------------------------------------------------------------------
// Kernel: out = pooled + (h1 @ w2 + b2)   M=384 N=1024 K=4096  (WMMA f32)
// ---------------------------------------------------------------------------
__global__ __launch_bounds__(256) void k_mlp2(const float* __restrict__ h1,
                                              const float* __restrict__ w2,
                                              const float* __restrict__ b2,
                                              const float* __restrict__ pooled,
                                              float* __restrict__ outp) {
  const int mt   = blockIdx.x;             // 0..23
  const int wave = threadIdx.x >> 5;
  const int lane = threadIdx.x & 31;
  const int half = lane & 15;
  const int m0   = mt * 16;
  const int n0   = (blockIdx.y * 8 + wave) * 16 + half;
  const int khalf = (lane >> 4) * 2;

  __shared__ __align__(16) float As[16 * ASTRIDE];

  v8f acc = {};
  for (int kc = 0; kc < HID; kc += 512) {
    __syncthreads();
    for (int j = threadIdx.x; j < 2048; j += 256) {
      const int row = j >> 7;
      const int c4  = (j & 127) << 2;
      ASYNC_CP16(As + row * ASTRIDE + c4,
                 h1 + (size_t)(m0 + row) * HID + kc + c4);
    }
    ASYNC_WAIT();
    __syncthreads();

    const float* arp = As + (size_t)half * ASTRIDE + khalf;
    const float* brp = w2 + (size_t)(kc + khalf) * DIMD + n0;
    for (int k = 0; k < 512; k += 4) {
      v2f a = *(const v2f*)(arp + k);
      v2f bb;
      bb.x = brp[(size_t)k * DIMD];
      bb.y = brp[(size_t)(k + 1) * DIMD];
      acc = wmma4(a, bb, acc);
    }
  }

  const float bias = b2[n0];
  const int mrow = m0 + 8 * (lane >> 4);
#pragma unroll
  for (int i = 0; i < 8; ++i) {
    size_t idx = (size_t)(mrow + i) * DIMD + n0;
    outp[idx] = acc[i] + bias + pooled[idx];
  }
}

// ---------------------------------------------------------------------------
extern "C" void kernel_launch(void* const* d_in, const int* in_sizes, int n_in,
                              void* d_out, int out_size, void* d_ws, size_t ws_size,
                              hipStream_t stream) {
  const float* x  = (const float*)d_in[0];   // [64, 4096, 1024]
  const float* q  = (const float*)d_in[1];   // [6, 1024]
  const float* g  = (const float*)d_in[2];   // [1024]
  const float* be = (const float*)d_in[3];   // [1024]
  const float* w1 = (const float*)d_in[4];   // [1024, 4096]
  const float* b1 = (const float*)d_in[5];   // [4096]
  const float* w2 = (const float*)d_in[6];   // [4096, 1024]
  const float* b2 = (const float*)d_in[7];   // [1024]
  float* out = (float*)d_out;                // [64*6, 1024]

  float* ws     = (float*)d_ws;
  float* pacc   = ws;                                        // 64*8*6*1024
  float* pm     = pacc + (size_t)NB * NSEG * NTOK * DIMD;    // 64*8*16
  float* pl     = pm   + (size_t)NB * NSEG * 16;             // 64*8*16
  float* pooled = pl   + (size_t)NB * NSEG * 16;             // 384*1024
  float* hn     = pooled + (size_t)NB * NTOK * DIMD;         // 384*1024
  float* h1     = hn     + (size_t)NB * NTOK * DIMD;         // 384*4096

  k_attnpool<<<dim3(NB, NSEG), 256, 0, stream>>>(x, q, pacc, pm, pl);
  k_combine <<<NB * NTOK,      256, 0, stream>>>(pacc, pm, pl, pooled);
  k_ln      <<<NB * NTOK,      256, 0, stream>>>(pooled, g, be, hn);
  k_mlp1    <<<dim3(24, 32),   256, 0, stream>>>(hn, w1, b1, h1);
  k_mlp2    <<<dim3(24, 8),    256, 0, stream>>>(h1, w2, b2, pooled, out);
}